// SAGEStage3_ReduceSum_47596827574313
// MI455X (gfx1250) — compile-verified
//
#include <hip/hip_runtime.h>

typedef __attribute__((ext_vector_type(4))) float v4f;

#define THREADS 256

// ---------------------------------------------------------------------------
// Kernel 1: zero the output accumulator (N*64 floats) with 16B stores.
// Regular-temporal stores on purpose: these lines are about to be hammered by
// L2-resident atomics, so we WANT them dirty/resident in L2.
// ---------------------------------------------------------------------------
__global__ void __launch_bounds__(THREADS)
zero_f4_kernel(v4f* __restrict__ out, int n4) {
    int i = blockIdx.x * blockDim.x + threadIdx.x;
    if (i < n4) {
        out[i] = (v4f){0.0f, 0.0f, 0.0f, 0.0f};
    }
}

// ---------------------------------------------------------------------------
// Kernel 2: scatter-add. Thread t handles 4-float (1/16-row) chunk q = t&15 of edge
// e = t>>4. 16 consecutive lanes cover one 256B edge row (coalesced b128
// loads, NT hint since messages stream through exactly once). The 4
// accumulations use hardware global_atomic_add_f32 (non-returning, STOREcnt)
// with immediate byte offsets — the 12.8MB output is fully L2-resident on a
// 192MB-L2 part, so these are fast L2 RMWs with ~16 avg updates per node.
// ---------------------------------------------------------------------------
__global__ void __launch_bounds__(THREADS)
scatter_add_f32_kernel(const float* __restrict__ messages,
                       const int* __restrict__ targets,
                       float* __restrict__ out,
                       int num_edges) {
    int t = blockIdx.x * blockDim.x + threadIdx.x;
    int e = t >> 4;
    int q = t & 15;
    if (e >= num_edges) return;

    int tgt = targets[e];  // 16 lanes share this 4B load -> broadcast in MC

    const v4f* mrow = (const v4f*)messages + ((size_t)e * 16 + (size_t)q);
    v4f m = __builtin_nontemporal_load(mrow);  // global_load_b128 th:NT

    float* dst = out + ((size_t)tgt * 64 + (size_t)(q * 4));

    asm volatile(
        "global_atomic_add_f32 %0, %1, off\n\t"
        "global_atomic_add_f32 %0, %2, off offset:4\n\t"
        "global_atomic_add_f32 %0, %3, off offset:8\n\t"
        "global_atomic_add_f32 %0, %4, off offset:12"
        :
        : "v"(dst), "v"(m.x), "v"(m.y), "v"(m.z), "v"(m.w)
        : "memory");
}

// ---------------------------------------------------------------------------
// d_in[0]: messages  float32 [E, 64]
// d_in[1]: edge_index int32  [2, E]   (row 1 = scatter targets)
// d_in[2]: num_nodes  (scalar on device; N recovered from out_size/64)
// d_out  : float32 [N, 64]
// ---------------------------------------------------------------------------
extern "C" void kernel_launch(void* const* d_in, const int* in_sizes, int n_in,
                              void* d_out, int out_size, void* d_ws, size_t ws_size,
                              hipStream_t stream) {
    (void)n_in; (void)d_ws; (void)ws_size;

    const float* messages = (const float*)d_in[0];
    const int* edge_index = (const int*)d_in[1];
    const int num_edges = in_sizes[1] / 2;
    const int* targets = edge_index + num_edges;  // row 1 of (2, E)
    float* out = (float*)d_out;

    // Zero the accumulator (out_size = N*64, always a multiple of 4).
    int n4 = out_size / 4;
    int zblocks = (n4 + THREADS - 1) / THREADS;
    zero_f4_kernel<<<zblocks, THREADS, 0, stream>>>((v4f*)out, n4);

    // Scatter-add: 16 threads per edge.
    long long total = (long long)num_edges * 16;
    int sblocks = (int)((total + THREADS - 1) / THREADS);
    scatter_add_f32_kernel<<<sblocks, THREADS, 0, stream>>>(messages, targets,
                                                            out, num_edges);
}